// LSTM_55817394979488
// MI455X (gfx1250) — compile-verified
//
#include <hip/hip_runtime.h>

// LSTM: T=262144 steps, INPUT=3, HIDDEN=32, OUT=1.
// Phase 1: single-wave32 sequential scan (latency-bound recurrence), weights in
//          VGPRs, h broadcast through LDS, writes hs[T,32] to d_ws.
// Phase 2: output projection hs @ W_lin^T + b via V_WMMA_F32_16X16X4_F32 with
//          W_lin replicated across all A rows (only D row 0 consumed) so every
//          lane operand is unconditionally live -> zero exec-mask branches.

#define TT   262144
#define HID  32

typedef float v2f __attribute__((ext_vector_type(2)));
typedef float v8f __attribute__((ext_vector_type(8)));

__device__ __forceinline__ float fast_rcp(float x) {
#if __has_builtin(__builtin_amdgcn_rcpf)
    return __builtin_amdgcn_rcpf(x);
#else
    return 1.0f / x;
#endif
}

__device__ __forceinline__ float fast_tanh(float x) {
#if __has_builtin(__builtin_amdgcn_tanhf)
    return __builtin_amdgcn_tanhf(x);     // CDNA5 v_tanh_f32 trans op
#else
    float e = __expf(2.0f * x);
    return 1.0f - 2.0f * fast_rcp(e + 1.0f);
#endif
}

__device__ __forceinline__ float fast_sigmoid(float x) {
    // sigmoid(x) = 0.5 * tanh(x/2) + 0.5  -> one trans op
    return fmaf(0.5f, fast_tanh(0.5f * x), 0.5f);
}

// ---------------------------------------------------------------------------
// Phase 1: sequential LSTM scan, one wave32.
// Lane l owns gate rows {l, 32+l, 64+l, 96+l} (i,f,g,o for hidden unit l), so
// the (c,h) update is fully lane-local; only h needs cross-lane broadcast.
// ---------------------------------------------------------------------------
__global__ __launch_bounds__(32) void lstm_scan(
    const float* __restrict__ x,      // [T,3]
    const float* __restrict__ W_ih,   // [128,3]
    const float* __restrict__ W_hh,   // [128,32]
    const float* __restrict__ b_ih,   // [128]
    const float* __restrict__ b_hh,   // [128]
    float* __restrict__ hs)           // [T,32]  (d_ws)
{
    const int l = threadIdx.x;        // 0..31
    __shared__ __align__(16) float sh_h[HID];

    // Recurrent weights: 4 rows of 32 f32 per lane, register-resident.
    float4 wi[8], wf[8], wg[8], wo[8];
    const float4* W4 = reinterpret_cast<const float4*>(W_hh);
#pragma unroll
    for (int k = 0; k < 8; ++k) {
        wi[k] = W4[(l      ) * 8 + k];
        wf[k] = W4[(l + 32 ) * 8 + k];
        wg[k] = W4[(l + 64 ) * 8 + k];
        wo[k] = W4[(l + 96 ) * 8 + k];
    }
    const float ui0 = W_ih[(l      )*3+0], ui1 = W_ih[(l      )*3+1], ui2 = W_ih[(l      )*3+2];
    const float uf0 = W_ih[(l + 32 )*3+0], uf1 = W_ih[(l + 32 )*3+1], uf2 = W_ih[(l + 32 )*3+2];
    const float ug0 = W_ih[(l + 64 )*3+0], ug1 = W_ih[(l + 64 )*3+1], ug2 = W_ih[(l + 64 )*3+2];
    const float uo0 = W_ih[(l + 96 )*3+0], uo1 = W_ih[(l + 96 )*3+1], uo2 = W_ih[(l + 96 )*3+2];
    const float bi = b_ih[l      ] + b_hh[l      ];
    const float bf = b_ih[l + 32 ] + b_hh[l + 32 ];
    const float bg = b_ih[l + 64 ] + b_hh[l + 64 ];
    const float bo = b_ih[l + 96 ] + b_hh[l + 96 ];

    float c = 0.0f;
    sh_h[l] = 0.0f;
    __syncthreads();

    for (int t = 0; t < TT; ++t) {
        // Prefetch the x stream ahead (tiny, L2-resident).
        int tp = (t + 32 < TT) ? (t + 32) : (TT - 1);
        __builtin_prefetch(&x[3 * tp], 0, 0);

        // Input contribution (fused; avoids a 134MB xg round-trip to HBM).
        const float x0 = x[3*t + 0], x1 = x[3*t + 1], x2 = x[3*t + 2];
        float ai = fmaf(ui2, x2, fmaf(ui1, x1, fmaf(ui0, x0, bi)));
        float af = fmaf(uf2, x2, fmaf(uf1, x1, fmaf(uf0, x0, bf)));
        float ag = fmaf(ug2, x2, fmaf(ug1, x1, fmaf(ug0, x0, bg)));
        float ao = fmaf(uo2, x2, fmaf(uo1, x1, fmaf(uo0, x0, bo)));

        // Broadcast-read h from LDS (same address across lanes: conflict-free).
        // Lockstep single wave: these ds_loads precede the ds_store below in
        // program order, and per-wave LDS ops complete in order -> no barrier
        // needed between read(old h) and write(new h).
        float4 hv[8];
        const float4* H4 = reinterpret_cast<const float4*>(sh_h);
#pragma unroll
        for (int k = 0; k < 8; ++k) hv[k] = H4[k];

        // 4x length-32 dot products: 128 FMAs, 4 independent chains for ILP.
#pragma unroll
        for (int k = 0; k < 8; ++k) {
            ai = fmaf(wi[k].x, hv[k].x, ai); ai = fmaf(wi[k].y, hv[k].y, ai);
            ai = fmaf(wi[k].z, hv[k].z, ai); ai = fmaf(wi[k].w, hv[k].w, ai);
            af = fmaf(wf[k].x, hv[k].x, af); af = fmaf(wf[k].y, hv[k].y, af);
            af = fmaf(wf[k].z, hv[k].z, af); af = fmaf(wf[k].w, hv[k].w, af);
            ag = fmaf(wg[k].x, hv[k].x, ag); ag = fmaf(wg[k].y, hv[k].y, ag);
            ag = fmaf(wg[k].z, hv[k].z, ag); ag = fmaf(wg[k].w, hv[k].w, ag);
            ao = fmaf(wo[k].x, hv[k].x, ao); ao = fmaf(wo[k].y, hv[k].y, ao);
            ao = fmaf(wo[k].z, hv[k].z, ao); ao = fmaf(wo[k].w, hv[k].w, ao);
        }

        const float gi = fast_sigmoid(ai);
        const float gf = fast_sigmoid(af);
        const float gg = fast_tanh(ag);
        const float go = fast_sigmoid(ao);
        c = fmaf(gf, c, gi * gg);
        const float h = go * fast_tanh(c);

        hs[t * HID + l] = h;          // coalesced 128B store, fire-and-forget
        sh_h[l] = h;
        __syncthreads();              // new h visible before next iteration
    }
}

// ---------------------------------------------------------------------------
// Phase 2: out[t] = dot(W_lin, hs[t]) + b_lin via V_WMMA_F32_16X16X4_F32.
// Roles: A = W_lin replicated into every row (16x4, lane-uniform per half:
// lanes 0-15 hold K{0,1}, lanes 16-31 hold K{2,3} -> selects between scalar-
// loaded values, no masking). B = hs tile with 16 timesteps as the N columns:
// B[k,n] = hs[t0+n, k0+k], VGPR0 = rows K{0|1}, VGPR1 = rows K{2|3}.
// D row 0 (= the GEMV result for 16 timesteps) lives in acc[0] of lanes 0-15
// -> single coalesced 64B store. No exec-mask changes anywhere near WMMA.
// ---------------------------------------------------------------------------
__global__ __launch_bounds__(256) void lstm_proj_wmma(
    const float* __restrict__ hs,     // [T,32]
    const float* __restrict__ W_lin,  // [1,32]
    const float* __restrict__ b_lin,  // [1]
    float* __restrict__ out)          // [T,1]
{
    const int lane = threadIdx.x & 31;
    const int half = lane >> 4;       // which K-pair this lane carries
    const int m    = lane & 15;       // N column (timestep within tile)
    const int wave = threadIdx.x >> 5;
    const int t0   = (blockIdx.x * 8 + wave) * 16;
    const float bl = b_lin[0];

    const float* __restrict__ hrow = hs + (size_t)(t0 + m) * HID;

    v8f acc = {};
#pragma unroll
    for (int kt = 0; kt < 8; ++kt) {
        const int k0 = kt * 4;
        // A: every row = W_lin[k0..k0+3]; uniform scalar loads + one select.
        v2f a;
        a.x = W_lin[k0 + 2 * half + 0];
        a.y = W_lin[k0 + 2 * half + 1];
        // B: lane (half,m) supplies B[half + {0,2}, m] = hs[t0+m, k0+half+{0,2}].
        v2f b;
        b.x = hrow[k0 + half + 0];
        b.y = hrow[k0 + half + 2];
        acc = __builtin_amdgcn_wmma_f32_16x16x4_f32(
            /*neg_a=*/false, a, /*neg_b=*/false, b,
            /*c_mod=*/(short)0, acc, /*reuse_a=*/false, /*reuse_b=*/false);
    }

    if (half == 0) {                  // D row 0: lanes 0-15, VGPR0
        out[t0 + m] = acc[0] + bl;    // coalesced 64B store
    }
}

extern "C" void kernel_launch(void* const* d_in, const int* in_sizes, int n_in,
                              void* d_out, int out_size, void* d_ws, size_t ws_size,
                              hipStream_t stream) {
    const float* x     = (const float*)d_in[0];
    const float* W_ih  = (const float*)d_in[1];
    const float* W_hh  = (const float*)d_in[2];
    const float* b_ih  = (const float*)d_in[3];
    const float* b_hh  = (const float*)d_in[4];
    const float* W_lin = (const float*)d_in[5];
    const float* b_lin = (const float*)d_in[6];
    float* out = (float*)d_out;
    float* hs  = (float*)d_ws;   // TT*HID*4 = 33.5 MB scratch

    lstm_scan<<<1, 32, 0, stream>>>(x, W_ih, W_hh, b_ih, b_hh, hs);
    lstm_proj_wmma<<<TT / 16 / 8, 256, 0, stream>>>(hs, W_lin, b_lin, out);
}